// Deepgmd_53704271069553
// MI455X (gfx1250) — compile-verified
//
#include <hip/hip_runtime.h>
#include <hip/hip_bf16.h>
#include <stdint.h>

// ---------------------------------------------------------------------------
// GraphSAGE-GCN (3 layers) for MI455X / gfx1250 (wave32).
//   agg = (sum_{src->dst} h[src] + h[dst]) * 1/(deg+1);  y = relu(agg @ W^T)
// Strategy: build CSR once per launch (int atomics), atomic-free per-node
// gather aggregation (wave per node), fp32 WMMA 16x16x4 GEMM + fused ReLU.
// ---------------------------------------------------------------------------

typedef __attribute__((ext_vector_type(2))) float v2f;
typedef __attribute__((ext_vector_type(8))) float v8f;

#define NN 50000      // nodes
#define NE 800000     // edges
#define DIN 128
#define DH  128
#define DZ  64

// ---------------- CSR construction ----------------

__global__ void k_zero_u32(unsigned* __restrict__ p, int n) {
  int i = blockIdx.x * blockDim.x + threadIdx.x;
  if (i < n) p[i] = 0u;
}

__global__ void k_count(const int* __restrict__ dst, unsigned* __restrict__ deg, int E) {
  int i = blockIdx.x * blockDim.x + threadIdx.x;
  if (i < E) atomicAdd(&deg[dst[i]], 1u);
}

// Single-block exclusive scan over NN degrees; also emits inv = 1/(deg+1).
__global__ void __launch_bounds__(1024) k_scan(const unsigned* __restrict__ deg,
                                               int* __restrict__ offs,
                                               float* __restrict__ inv, int n) {
  __shared__ unsigned sh[1024];
  const int t = threadIdx.x;
  unsigned carry = 0u;
  for (int base = 0; base < n; base += 1024) {
    const int i = base + t;
    const unsigned d = (i < n) ? deg[i] : 0u;
    sh[t] = d;
    __syncthreads();
    for (int off = 1; off < 1024; off <<= 1) {
      const unsigned add = (t >= off) ? sh[t - off] : 0u;
      __syncthreads();
      sh[t] += add;
      __syncthreads();
    }
    if (i < n) {
      offs[i] = (int)(carry + sh[t] - d);            // exclusive prefix
      inv[i]  = 1.0f / (float)(d + 1u);
    }
    carry += sh[1023];
    __syncthreads();
  }
  if (t == 0) offs[n] = (int)carry;
}

__global__ void k_fill(const int* __restrict__ src, const int* __restrict__ dst,
                       const int* __restrict__ offs, unsigned* __restrict__ cur,
                       int* __restrict__ csr, int E) {
  int i = blockIdx.x * blockDim.x + threadIdx.x;
  if (i < E) {
    const int d = dst[i];
    const unsigned p = atomicAdd(&cur[d], 1u);
    csr[offs[d] + (int)p] = src[i];
  }
}

// ---------------- aggregation: one wave32 per node, float4 per lane --------

__global__ void __launch_bounds__(256) k_agg(const float* __restrict__ h,
                                             const int* __restrict__ csr,
                                             const int* __restrict__ offs,
                                             const float* __restrict__ inv,
                                             float* __restrict__ out) {
  const int wv   = (int)((blockIdx.x * blockDim.x + threadIdx.x) >> 5);
  const int lane = (int)(threadIdx.x & 31);
  if (wv >= NN) return;                       // wave-uniform
  const int beg = offs[wv];
  const int end = offs[wv + 1];
  const size_t fo = (size_t)lane * 4;

  const float4 self = *(const float4*)(h + (size_t)wv * DIN + fo);
  float sx = self.x, sy = self.y, sz = self.z, sw = self.w;

  for (int j = beg; j < end; ++j) {
    const int s = csr[j];
    if (j + 1 < end) {
      // hide L2 gather latency for the next neighbor row
      __builtin_prefetch(h + (size_t)csr[j + 1] * DIN + fo, 0, 3);
    }
    const float4 v = *(const float4*)(h + (size_t)s * DIN + fo);
    sx += v.x; sy += v.y; sz += v.z; sw += v.w;
  }
  const float sc = inv[wv];
  float4 r;
  r.x = sx * sc; r.y = sy * sc; r.z = sz * sc; r.w = sw * sc;
  *(float4*)(out + (size_t)wv * DIN + fo) = r;
}

// ---------------- fp32 WMMA GEMM + fused ReLU ------------------------------
// Y[m, n] = relu( sum_k A[m, k] * W[n, k] ),  one wave per 16x16 output tile.
// V_WMMA_F32_16X16X4_F32 fragment layout (ISA 7.12.2):
//   A (16x4):  lanes 0-15 -> M=lane, VGPR0=K0, VGPR1=K1; lanes 16-31 -> K2, K3
//   B (4x16):  lanes 0-15 -> N=lane, VGPR0=K0, VGPR1=K1; lanes 16-31 -> K2, K3
//   C/D:       VGPR r: lanes 0-15 -> row r, lanes 16-31 -> row r+8; col = lane%16

__global__ void __launch_bounds__(256) k_gemm_relu(const float* __restrict__ A,
                                                   const float* __restrict__ W,
                                                   float* __restrict__ Y,
                                                   int K, int Nout,
                                                   int ntiles, int ntot) {
  const int wv   = (int)((blockIdx.x * blockDim.x + threadIdx.x) >> 5);
  const int lane = (int)(threadIdx.x & 31);
  if (wv >= ntot) return;                     // wave-uniform; EXEC stays full
  const int mt = wv / ntiles;
  const int nt = wv - mt * ntiles;
  const int half = lane >> 4;                 // 0: K0/K1, 1: K2/K3
  const int lr   = lane & 15;

  const float* __restrict__ ar = A + (size_t)(mt * 16 + lr) * K;  // A row m
  const float* __restrict__ br = W + (size_t)(nt * 16 + lr) * K;  // W row n (== B col n)

  v8f c = {};
#pragma unroll 8
  for (int k0 = 0; k0 < K; k0 += 4) {
    const int ka = k0 + (half << 1);
    const v2f a = *(const v2f*)(ar + ka);     // 8B-aligned (ka even)
    const v2f b = *(const v2f*)(br + ka);
    c = __builtin_amdgcn_wmma_f32_16x16x4_f32(
        /*neg_a=*/false, a, /*neg_b=*/false, b,
        /*c_mod=*/(short)0, c, /*reuse_a=*/false, /*reuse_b=*/false);
  }

  float* __restrict__ yb = Y + (size_t)(mt * 16 + half * 8) * Nout + nt * 16 + lr;
#pragma unroll
  for (int r = 0; r < 8; ++r) {
    const float v = c[r];
    yb[(size_t)r * Nout] = v > 0.0f ? v : 0.0f;
  }
}

// ---------------- driver ---------------------------------------------------

extern "C" void kernel_launch(void* const* d_in, const int* in_sizes, int n_in,
                              void* d_out, int out_size, void* d_ws, size_t ws_size,
                              hipStream_t stream) {
  (void)in_sizes; (void)n_in; (void)out_size; (void)ws_size;

  const float* x  = (const float*)d_in[0];
  const int*   ei = (const int*)d_in[1];        // (2, NE) int32
  const float* W1 = (const float*)d_in[2];      // (128,128)
  const float* W2 = (const float*)d_in[3];      // (128,128)
  const float* W3 = (const float*)d_in[4];      // (64,128)
  const int* src = ei;
  const int* dst = ei + NE;

  // workspace carve-out (256B aligned slots)
  char* ws = (char*)d_ws;
  size_t off = 0;
  auto carve = [&](size_t bytes) -> void* {
    void* p = ws + off;
    off += (bytes + 255) & ~(size_t)255;
    return p;
  };
  unsigned* deg  = (unsigned*)carve((size_t)NN * 4);
  unsigned* cur  = (unsigned*)carve((size_t)NN * 4);
  int*      offs = (int*)carve(((size_t)NN + 1) * 4);
  float*    inv  = (float*)carve((size_t)NN * 4);
  int*      csr  = (int*)carve((size_t)NE * 4);
  float*    agg  = (float*)carve((size_t)NN * DIN * 4);   // 25.6 MB
  float*    hbuf = (float*)carve((size_t)NN * DH * 4);    // 25.6 MB

  const int T = 256;
  const int nB  = (NN + T - 1) / T;
  const int eB  = (NE + T - 1) / T;
  const int agB = (NN * 32 + T - 1) / T;      // one wave32 per node

  // --- CSR build (once per launch, deterministic work) ---
  k_zero_u32<<<nB, T, 0, stream>>>(deg, NN);
  k_zero_u32<<<nB, T, 0, stream>>>(cur, NN);
  k_count<<<eB, T, 0, stream>>>(dst, deg, NE);
  k_scan<<<1, 1024, 0, stream>>>(deg, offs, inv, NN);
  k_fill<<<eB, T, 0, stream>>>(src, dst, offs, cur, csr, NE);

  // --- layer 1: agg(x) -> GEMM W1 -> relu -> hbuf ---
  k_agg<<<agB, T, 0, stream>>>(x, csr, offs, inv, agg);
  {
    const int ntiles = DH / 16, ntot = (NN / 16) * ntiles;
    k_gemm_relu<<<(ntot * 32 + T - 1) / T, T, 0, stream>>>(agg, W1, hbuf, DIN, DH, ntiles, ntot);
  }
  // --- layer 2: agg(h1) -> GEMM W2 -> relu -> hbuf ---
  k_agg<<<agB, T, 0, stream>>>(hbuf, csr, offs, inv, agg);
  {
    const int ntiles = DH / 16, ntot = (NN / 16) * ntiles;
    k_gemm_relu<<<(ntot * 32 + T - 1) / T, T, 0, stream>>>(agg, W2, hbuf, DH, DH, ntiles, ntot);
  }
  // --- layer 3: agg(h2) -> GEMM W3 -> relu -> d_out ---
  k_agg<<<agB, T, 0, stream>>>(hbuf, csr, offs, inv, agg);
  {
    const int ntiles = DZ / 16, ntot = (NN / 16) * ntiles;
    k_gemm_relu<<<(ntot * 32 + T - 1) / T, T, 0, stream>>>(agg, W3, (float*)d_out, DH, DZ, ntiles, ntot);
  }
}